// network_5299989643808
// MI455X (gfx1250) — compile-verified
//
#include <hip/hip_runtime.h>
#include <stdint.h>

// ---------------------------------------------------------------------------
// Problem constants (match reference)
// ---------------------------------------------------------------------------
#define ZN      64
#define BATCH   2048
#define HDIM    4
#define NOPT    4
#define LN_EPS  1e-5f
#define LOG2E   1.4426950408889634f
#define LN2     0.6931471805599453f
#define GROUPS  (BATCH * ZN * ZN)       // 8,388,608 (b,z,zp) cells, 4 floats each

// LDS layout, dword offsets. Row strides padded to ODD dword counts so that
// 32 lanes (distinct zp / distinct b) never collide on the 64 LDS banks.
#define S_W1    0       // 64 rows * stride 9  (8 data + 1 pad)
#define S_B1    576     // 64 rows * stride 5  (4 data + 1 pad)
#define S_G1    896
#define S_BE1   1216
#define S_W2    1536    // 64 rows * stride 17 (16 data + 1 pad)
#define S_B2    2624
#define S_G2    2944
#define S_BE2   3264
#define S_WPI   3584    // stride 17
#define S_BPI   4672
#define S_WXI   4992    // stride 17
#define S_BXI   6080
#define S_NT    6400    // 64 rows * stride 65 (64 data + 1 pad)
#define SMEM_DWORDS 10560   // 42.2 KB of the WGP's 320 KB

typedef __attribute__((ext_vector_type(4))) float        f32x4;
typedef __attribute__((ext_vector_type(4))) unsigned int tdm_u32x4;
typedef __attribute__((ext_vector_type(8))) int          tdm_i32x8;
typedef __attribute__((ext_vector_type(4))) int          tdm_i32x4;

// ---------------------------------------------------------------------------
// Fast math on CDNA5 transcendental pipes
// ---------------------------------------------------------------------------
__device__ __forceinline__ float fexp2(float x) { return __builtin_amdgcn_exp2f(x); }
__device__ __forceinline__ float flog2(float x) { return __builtin_amdgcn_logf(x);  }
__device__ __forceinline__ float frcp (float x) { return __builtin_amdgcn_rcpf(x);  }
__device__ __forceinline__ float frsq (float x) { return __builtin_amdgcn_rsqf(x);  }

__device__ __forceinline__ float ftanh(float x) {
    // tanh(x) = 1 - 2/(1 + e^{2x});  e^{2x} = exp2(2*log2e*x)
    float e = fexp2(x * (2.0f * LOG2E));
    return 1.0f - 2.0f * frcp(1.0f + e);
}
__device__ __forceinline__ float fsigmoid(float t) {
    return frcp(1.0f + fexp2(-t * LOG2E));
}
__device__ __forceinline__ void ln4(float h[4], const float g[4], const float be[4]) {
    float m  = (h[0] + h[1] + h[2] + h[3]) * 0.25f;
    float d0 = h[0] - m, d1 = h[1] - m, d2 = h[2] - m, d3 = h[3] - m;
    float v  = (d0 * d0 + d1 * d1 + d2 * d2 + d3 * d3) * 0.25f;
    float r  = frsq(v + LN_EPS);
    h[0] = fmaf(d0 * r, g[0], be[0]);
    h[1] = fmaf(d1 * r, g[1], be[1]);
    h[2] = fmaf(d2 * r, g[2], be[2]);
    h[3] = fmaf(d3 * r, g[3], be[3]);
}

// ---------------------------------------------------------------------------
// TDM: stage 64 rows of `row_dw` dwords (contiguous in HBM) into LDS with one
// pad dword inserted after every row -> padded (odd) stride, conflict-free.
// pad_interval codes (dwords before pad): 1 -> 4, 2 -> 8, 3 -> 16, 5 -> 64.
// ---------------------------------------------------------------------------
#if __has_builtin(__builtin_amdgcn_tensor_load_to_lds)
#define HAVE_TDM 1
__device__ __forceinline__ void tdm_stage(const float* gsrc, unsigned lds_byte,
                                          unsigned row_dw, unsigned pad_code) {
    unsigned long long ga = (unsigned long long)(uintptr_t)gsrc;
    tdm_u32x4 g0;
    g0[0] = 1u;                                  // count=1 (valid user D#)
    g0[1] = lds_byte;                            // lds_addr (bytes, wave-relative)
    g0[2] = (unsigned)ga;                        // global_addr[31:0]
    g0[3] = (unsigned)(ga >> 32) | (2u << 30);   // global_addr[56:32] | type=2
    tdm_i32x8 g1;
    g1[0] = (int)((2u << 16)                     // data_size = 4 bytes
                | (1u << 20)                     // pad_enable
                | (pad_code << 22));             // pad_interval; pad_amount=0 (1 dword)
    g1[1] = (int)(row_dw << 16);                          // tensor_dim0[15:0]
    g1[2] = (int)((row_dw >> 16) | ((unsigned)ZN << 16)); // dim0[31:16] | tensor_dim1=64
    g1[3] = (int)(row_dw << 16);                          // dim1[31:16]=0 | tile_dim0
    g1[4] = (int)ZN;                                      // tile_dim1=64, tile_dim2=0
    g1[5] = (int)row_dw;                                  // tensor_dim0_stride (rows contiguous)
    g1[6] = 0;
    g1[7] = 0;
    tdm_i32x4 z4 = (tdm_i32x4)0;
#if defined(__clang_major__) && __clang_major__ >= 23
    tdm_i32x8 z8 = (tdm_i32x8)0;
    __builtin_amdgcn_tensor_load_to_lds(g0, g1, z4, z4, z8, 0);
#else
    __builtin_amdgcn_tensor_load_to_lds(g0, g1, z4, z4, 0);
#endif
}
#else
#define HAVE_TDM 0
__device__ __forceinline__ void coop_stage(float* smem, const float* gsrc,
                                           unsigned lds_dw, unsigned row_dw,
                                           unsigned tid) {
    unsigned total = ZN * row_dw;                // row_dw is a power of two
    for (unsigned i = tid; i < total; i += 256u) {
        unsigned r = i / row_dw, c = i - r * row_dw;
        smem[lds_dw + r * (row_dw + 1u) + c] = gsrc[i];
    }
}
#endif

// ---------------------------------------------------------------------------
// One block = one z, one tile of 64 batch rows. threads: zp = tid&63, 4 b-rows.
// Store-bound kernel: every wave writes 512 B contiguous per output tensor,
// and after TDM staging the inner loop touches global memory ONLY with stores.
// ---------------------------------------------------------------------------
__global__ __launch_bounds__(256, 1)
void pair_mlp_kernel(const float* __restrict__ nt,
                     const float* __restrict__ W1,  const float* __restrict__ b1,
                     const float* __restrict__ g1,  const float* __restrict__ be1,
                     const float* __restrict__ W2,  const float* __restrict__ b2,
                     const float* __restrict__ g2,  const float* __restrict__ be2,
                     const float* __restrict__ Wpi, const float* __restrict__ bpi,
                     const float* __restrict__ Wxi, const float* __restrict__ bxi,
                     float* __restrict__ out) {
    __shared__ float smem[SMEM_DWORDS];

    const unsigned tid  = threadIdx.x;
    const unsigned z    = blockIdx.x >> 5;     // 64 z values
    const unsigned bt   = blockIdx.x & 31u;    // 32 tiles of 64 batch rows
    const unsigned zp   = tid & 63u;
    const unsigned brow = tid >> 6;            // 0..3

    // ---- stage weights (z slice) + nt (b tile) into padded LDS via TDM ----
#if HAVE_TDM
    if (tid < 32u) {                           // one wave drives the TDM
        const unsigned smb = (unsigned)(uintptr_t)(void*)smem;  // LDS byte offset
        const unsigned zo8 = z * (ZN * 8), zo4 = z * (ZN * 4), zo16 = z * (ZN * 16);
        tdm_stage(W1  + zo8,        smb + S_W1  * 4,  8u, 2u);
        tdm_stage(b1  + zo4,        smb + S_B1  * 4,  4u, 1u);
        tdm_stage(g1  + zo4,        smb + S_G1  * 4,  4u, 1u);
        tdm_stage(be1 + zo4,        smb + S_BE1 * 4,  4u, 1u);
        tdm_stage(W2  + zo16,       smb + S_W2  * 4, 16u, 3u);
        tdm_stage(b2  + zo4,        smb + S_B2  * 4,  4u, 1u);
        tdm_stage(g2  + zo4,        smb + S_G2  * 4,  4u, 1u);
        tdm_stage(be2 + zo4,        smb + S_BE2 * 4,  4u, 1u);
        tdm_stage(Wpi + zo16,       smb + S_WPI * 4, 16u, 3u);
        tdm_stage(bpi + zo4,        smb + S_BPI * 4,  4u, 1u);
        tdm_stage(Wxi + zo16,       smb + S_WXI * 4, 16u, 3u);
        tdm_stage(bxi + zo4,        smb + S_BXI * 4,  4u, 1u);
        tdm_stage(nt  + bt * 4096u, smb + S_NT  * 4, 64u, 5u);  // 64x64 nt tile
        __builtin_amdgcn_s_wait_tensorcnt(0);
    }
#else
    {
        const unsigned zo8 = z * (ZN * 8), zo4 = z * (ZN * 4), zo16 = z * (ZN * 16);
        coop_stage(smem, W1  + zo8,        S_W1,   8u, tid);
        coop_stage(smem, b1  + zo4,        S_B1,   4u, tid);
        coop_stage(smem, g1  + zo4,        S_G1,   4u, tid);
        coop_stage(smem, be1 + zo4,        S_BE1,  4u, tid);
        coop_stage(smem, W2  + zo16,       S_W2,  16u, tid);
        coop_stage(smem, b2  + zo4,        S_B2,   4u, tid);
        coop_stage(smem, g2  + zo4,        S_G2,   4u, tid);
        coop_stage(smem, be2 + zo4,        S_BE2,  4u, tid);
        coop_stage(smem, Wpi + zo16,       S_WPI, 16u, tid);
        coop_stage(smem, bpi + zo4,        S_BPI,  4u, tid);
        coop_stage(smem, Wxi + zo16,       S_WXI, 16u, tid);
        coop_stage(smem, bxi + zo4,        S_BXI,  4u, tid);
        coop_stage(smem, nt  + bt * 4096u, S_NT,  64u, tid);
    }
#endif
    __syncthreads();

    // ---- hoist this zp's 88 weights into registers (conflict-free reads) --
    float w1v[8], b1v[4], g1v[4], be1v[4];
    float w2v[16], b2v[4], g2v[4], be2v[4];
    float wpiv[16], bpiv[4], wxiv[16], bxiv[4];
#pragma unroll
    for (int j = 0; j < 8;  ++j) w1v[j]  = smem[S_W1  + zp * 9  + j];
#pragma unroll
    for (int j = 0; j < 4;  ++j) {
        b1v[j]  = smem[S_B1  + zp * 5 + j];
        g1v[j]  = smem[S_G1  + zp * 5 + j];
        be1v[j] = smem[S_BE1 + zp * 5 + j];
        b2v[j]  = smem[S_B2  + zp * 5 + j];
        g2v[j]  = smem[S_G2  + zp * 5 + j];
        be2v[j] = smem[S_BE2 + zp * 5 + j];
        bpiv[j] = smem[S_BPI + zp * 5 + j];
        bxiv[j] = smem[S_BXI + zp * 5 + j];
    }
#pragma unroll
    for (int j = 0; j < 16; ++j) {
        w2v[j]  = smem[S_W2  + zp * 17 + j];
        wpiv[j] = smem[S_WPI + zp * 17 + j];
        wxiv[j] = smem[S_WXI + zp * 17 + j];
    }

    f32x4* __restrict__ opi = (f32x4*)out;
    f32x4* __restrict__ olp = opi + GROUPS;
    f32x4* __restrict__ oxi = olp + GROUPS;

#pragma unroll 4
    for (unsigned it = 0; it < 16u; ++it) {
        const unsigned ib = it * 4u + brow;           // 0..63 within the b tile
        const unsigned b  = bt * 64u + ib;
        const float x0 = smem[S_NT + ib * 65u + z];   // wave-uniform broadcast
        const float x1 = smem[S_NT + ib * 65u + zp];  // lane-varying, odd stride

        // layer 1: 2 -> 4, tanh, LN
        float h[4];
#pragma unroll
        for (int j = 0; j < 4; ++j)
            h[j] = ftanh(fmaf(x0, w1v[j], fmaf(x1, w1v[4 + j], b1v[j])));
        ln4(h, g1v, be1v);

        // layer 2: 4 -> 4, tanh, LN
        float h2[4];
#pragma unroll
        for (int j = 0; j < 4; ++j) {
            float a = b2v[j];
#pragma unroll
            for (int i = 0; i < 4; ++i) a = fmaf(h[i], w2v[i * 4 + j], a);
            h2[j] = ftanh(a);
        }
        ln4(h2, g2v, be2v);

        // pi head: logits -> stable log_softmax (all in-register, width 4)
        float pl[4];
#pragma unroll
        for (int o = 0; o < 4; ++o) {
            float a = bpiv[o];
#pragma unroll
            for (int k = 0; k < 4; ++k) a = fmaf(h2[k], wpiv[k * 4 + o], a);
            pl[o] = a;
        }
        float mx = fmaxf(fmaxf(pl[0], pl[1]), fmaxf(pl[2], pl[3]));
        float e0 = fexp2((pl[0] - mx) * LOG2E);
        float e1 = fexp2((pl[1] - mx) * LOG2E);
        float e2 = fexp2((pl[2] - mx) * LOG2E);
        float e3 = fexp2((pl[3] - mx) * LOG2E);
        float s    = e0 + e1 + e2 + e3;
        float invs = frcp(s);
        float lse  = mx + flog2(s) * LN2;

        f32x4 pi4 = { e0 * invs, e1 * invs, e2 * invs, e3 * invs };
        f32x4 lp4 = { pl[0] - lse, pl[1] - lse, pl[2] - lse, pl[3] - lse };

        // xi head: sigmoid
        f32x4 xi4;
#pragma unroll
        for (int o = 0; o < 4; ++o) {
            float a = bxiv[o];
#pragma unroll
            for (int k = 0; k < 4; ++k) a = fmaf(h2[k], wxiv[k * 4 + o], a);
            xi4[o] = fsigmoid(a);
        }

        // coalesced 128-bit NT stores: lanes (zp) -> consecutive 16B chunks
        const size_t gidx = ((size_t)b * ZN + z) * ZN + zp;
        __builtin_nontemporal_store(pi4, &opi[gidx]);
        __builtin_nontemporal_store(lp4, &olp[gidx]);
        __builtin_nontemporal_store(xi4, &oxi[gidx]);
    }
}

// ---------------------------------------------------------------------------
extern "C" void kernel_launch(void* const* d_in, const int* in_sizes, int n_in,
                              void* d_out, int out_size, void* d_ws, size_t ws_size,
                              hipStream_t stream) {
    (void)in_sizes; (void)n_in; (void)out_size; (void)d_ws; (void)ws_size;
    const float* nt  = (const float*)d_in[0];
    const float* W1  = (const float*)d_in[1];
    const float* b1  = (const float*)d_in[2];
    const float* g1  = (const float*)d_in[3];
    const float* be1 = (const float*)d_in[4];
    const float* W2  = (const float*)d_in[5];
    const float* b2  = (const float*)d_in[6];
    const float* g2  = (const float*)d_in[7];
    const float* be2 = (const float*)d_in[8];
    const float* Wpi = (const float*)d_in[9];
    const float* bpi = (const float*)d_in[10];
    const float* Wxi = (const float*)d_in[11];
    const float* bxi = (const float*)d_in[12];

    dim3 grid(ZN * (BATCH / 64));   // 2048 blocks: (z, b-tile)
    dim3 block(256);                // 8 wave32 per block
    pair_mlp_kernel<<<grid, block, 0, stream>>>(nt, W1, b1, g1, be1,
                                                W2, b2, g2, be2,
                                                Wpi, bpi, Wxi, bxi,
                                                (float*)d_out);
}